// MergeWindows_25220047962163
// MI455X (gfx1250) — compile-verified
//
#include <hip/hip_runtime.h>

// ---------------------------------------------------------------------------
// Problem constants (fixed by the reference harness)
// ---------------------------------------------------------------------------
#define IMG_H 1024
#define IMG_W 1024
#define N_CH  32
#define N_WIN 4
#define MPW   8           // channels per window
#define WIN_H 512
#define WIN_W 512
#define SLOT_DIM 64
#define SIM_THRESH 0.1f
#define PLANE (IMG_H * IMG_W)          // 1,048,576 pixels per channel plane
#define PLANE4 (PLANE / 4)             // in float4 units

typedef float v2f __attribute__((ext_vector_type(2)));
typedef float v8f __attribute__((ext_vector_type(8)));

// Workspace layout (int units):
//   [0..31]    edge_l
//   [32..63]   edge_r
//   [64..95]   edge_t
//   [96..127]  edge_b
//   [128..159] channel remap table
#define WS_EDGE_L 0
#define WS_EDGE_R 32
#define WS_EDGE_T 64
#define WS_EDGE_B 96
#define WS_MAP    128

__device__ __forceinline__ int argmax32(const float* __restrict__ masks, int off) {
    float best = masks[off];
    int idx = 0;
    #pragma unroll
    for (int c = 1; c < N_CH; ++c) {
        float v = masks[c * PLANE + off];
        if (v > best) { best = v; idx = c; }   // first max wins (jnp.argmax)
    }
    return idx;
}

// ---------------------------------------------------------------------------
// Kernel Z: zero the edge-flag scratch (harness does not re-poison between calls)
// ---------------------------------------------------------------------------
__global__ void mw_zero_kernel(int* __restrict__ ws) {
    ws[threadIdx.x] = 0;   // 128 threads cover the 4x32 edge flags
}

// ---------------------------------------------------------------------------
// Kernel B: per-window border edge flags. One block per window, 512 threads.
// Recomputes the border-pixel argmax directly (~1 MB of reads total).
// ---------------------------------------------------------------------------
__global__ void mw_edges_kernel(const float* __restrict__ masks,
                                const int* __restrict__ pad_left,
                                const int* __restrict__ pad_top,
                                int* __restrict__ ws) {
    const int wi = blockIdx.x;
    const int pl = pad_left[wi], pt = pad_top[wi];
    const int ys = pt > 0 ? pt : 0;
    const int ye = (pt + WIN_H) < IMG_H ? (pt + WIN_H) : IMG_H;
    const int xs = pl > 0 ? pl : 0;
    const int xe = (pl + WIN_W) < IMG_W ? (pl + WIN_W) : IMG_W;
    if (ys >= ye || xs >= xe) return;

    const int s = wi * MPW, e = s + MPW;
    const int t = threadIdx.x;

    if (t < ye - ys) {
        const int y = ys + t;
        int id = argmax32(masks, y * IMG_W + xs);
        if (id >= s && id < e) atomicOr(&ws[WS_EDGE_L + id], 1);
        id = argmax32(masks, y * IMG_W + (xe - 1));
        if (id >= s && id < e) atomicOr(&ws[WS_EDGE_R + id], 1);
    }
    if (t < xe - xs) {
        const int x = xs + t;
        int id = argmax32(masks, ys * IMG_W + x);
        if (id >= s && id < e) atomicOr(&ws[WS_EDGE_T + id], 1);
        id = argmax32(masks, (ye - 1) * IMG_W + x);
        if (id >= s && id < e) atomicOr(&ws[WS_EDGE_B + id], 1);
    }
}

// ---------------------------------------------------------------------------
// Kernel C: single wave32 block.
//   1) Normalize the 28 slot-feature rows into a zero-padded 32x64 LDS matrix S.
//   2) Gram = S * S^T via V_WMMA_F32_16X16X4_F32 (4 tiles x 16 K-chunks, exact fp32).
//   3) Lane 0 builds adjacency from pads (exact reference ordering) and runs the
//      sequential merge scan, emitting the 32-entry channel remap table.
// ---------------------------------------------------------------------------
__global__ void mw_gram_merge_kernel(const float* __restrict__ sf,
                                     const int* __restrict__ pad_left,
                                     const int* __restrict__ pad_top,
                                     const int* __restrict__ edges,
                                     int* __restrict__ map_out) {
    __shared__ float S[N_CH][SLOT_DIM];   // 8 KB
    __shared__ float G[N_CH][N_CH];       // 4 KB

    const int lane = threadIdx.x;         // 0..31, one wave

    // --- normalize rows (rows 28..31 zero-padded) ---
    if (lane < N_WIN * (MPW - 1)) {
        const float* row = sf + lane * SLOT_DIM;
        float ss = 0.f;
        #pragma unroll
        for (int k = 0; k < SLOT_DIM; ++k) { float v = row[k]; ss += v * v; }
        const float inv = 1.0f / (__builtin_sqrtf(ss) + 1e-8f);
        #pragma unroll
        for (int k = 0; k < SLOT_DIM; ++k) S[lane][k] = row[k] * inv;
    } else {
        #pragma unroll
        for (int k = 0; k < SLOT_DIM; ++k) S[lane][k] = 0.f;
    }
    __syncthreads();

    // --- 32x32 Gram via WMMA f32 16x16x4 ---
    // A 16x4 layout (ISA 7.12.2): lanes 0-15 hold K=0,1; lanes 16-31 hold K=2,3.
    // B = S^T shares the mirrored layout: B[k][n] = S[n][k].
    // C/D layout: VGPR v -> rows v (lanes 0-15) and v+8 (lanes 16-31).
    const int half = lane >> 4;           // 0 or 1
    const int lm   = lane & 15;
    #pragma unroll
    for (int tm = 0; tm < 2; ++tm) {
        #pragma unroll
        for (int tn = 0; tn < 2; ++tn) {
            v8f acc = {0.f, 0.f, 0.f, 0.f, 0.f, 0.f, 0.f, 0.f};
            const int mrow = tm * 16 + lm;
            const int nrow = tn * 16 + lm;
            #pragma unroll
            for (int k0 = 0; k0 < SLOT_DIM; k0 += 4) {
                const int ka = k0 + half * 2;
                v2f a, b;
                a[0] = S[mrow][ka]; a[1] = S[mrow][ka + 1];
                b[0] = S[nrow][ka]; b[1] = S[nrow][ka + 1];
                acc = __builtin_amdgcn_wmma_f32_16x16x4_f32(
                    /*neg_a=*/false, a, /*neg_b=*/false, b,
                    /*c_mod=*/(short)0, acc, /*reuse_a=*/false, /*reuse_b=*/false);
            }
            #pragma unroll
            for (int v = 0; v < 8; ++v)
                G[tm * 16 + half * 8 + v][tn * 16 + lm] = acc[v];
        }
    }
    __syncthreads();

    // --- sequential merge scan on lane 0 ---
    if (lane == 0) {
        int pl[N_WIN], pt[N_WIN];
        #pragma unroll
        for (int w = 0; w < N_WIN; ++w) { pl[w] = pad_left[w]; pt[w] = pad_top[w]; }

        // adjacency in the reference's exact append order
        int awi[12], awj[12]; bool ahz[12]; int na = 0;
        for (int i = 0; i < N_WIN; ++i) {
            for (int j = i + 1; j < N_WIN; ++j) {
                int dl = pl[i] - pl[j]; if (dl < 0) dl = -dl;
                int dt = pt[i] - pt[j]; if (dt < 0) dt = -dt;
                if (pt[i] == pt[j] && dl == WIN_W) {
                    if (pl[i] < pl[j]) { awi[na] = i; awj[na] = j; }
                    else               { awi[na] = j; awj[na] = i; }
                    ahz[na++] = true;
                }
                if (pl[i] == pl[j] && dt == WIN_H) {
                    if (pt[i] < pt[j]) { awi[na] = i; awj[na] = j; }
                    else               { awi[na] = j; awj[na] = i; }
                    ahz[na++] = false;
                }
            }
        }

        bool merged[N_CH];
        int  cmap[N_CH];
        #pragma unroll
        for (int c = 0; c < N_CH; ++c) { merged[c] = false; cmap[c] = c; }

        for (int a = 0; a < na; ++a) {
            const int wi = awi[a], wj = awj[a];
            const bool hz = ahz[a];
            const int si = wi * MPW, sj = wj * MPW;
            for (int ci = si + 1; ci < si + MPW; ++ci) {
                for (int cj = sj + 1; cj < sj + MPW; ++cj) {
                    const float sim = G[wi * (MPW - 1) + (ci - si - 1)]
                                       [wj * (MPW - 1) + (cj - sj - 1)];
                    const bool ok = hz
                        ? (edges[WS_EDGE_R + ci] && edges[WS_EDGE_L + cj])
                        : (edges[WS_EDGE_B + ci] && edges[WS_EDGE_T + cj]);
                    if (ok && sim > SIM_THRESH && !merged[ci] && !merged[cj]) {
                        const int keep = ci < cj ? ci : cj;
                        const int rem  = ci < cj ? cj : ci;
                        for (int d = 0; d < N_CH; ++d)
                            if (cmap[d] == rem) cmap[d] = keep;
                        merged[rem] = true;
                    }
                }
            }
        }
        for (int c = 0; c < N_CH; ++c) map_out[c] = cmap[c];
    }
}

// ---------------------------------------------------------------------------
// Kernel D: fused argmax -> remap -> one-hot write, float4 vectorized.
// 256 MB of streaming traffic; this is the bandwidth-bound bulk (~11 us ideal).
// ---------------------------------------------------------------------------
__global__ void mw_onehot_kernel(const float* __restrict__ masks,
                                 const int* __restrict__ cmap,
                                 float* __restrict__ out) {
    __shared__ int smap[N_CH];
    if (threadIdx.x < N_CH) smap[threadIdx.x] = cmap[threadIdx.x];
    __syncthreads();

    const int g = blockIdx.x * blockDim.x + threadIdx.x;   // float4 group index
    const float4* mp = (const float4*)masks;

    float4 best = mp[g];
    int ix = 0, iy = 0, iz = 0, iw = 0;
    #pragma unroll
    for (int c = 1; c < N_CH; ++c) {
        const float4 v = mp[c * PLANE4 + g];
        if (v.x > best.x) { best.x = v.x; ix = c; }
        if (v.y > best.y) { best.y = v.y; iy = c; }
        if (v.z > best.z) { best.z = v.z; iz = c; }
        if (v.w > best.w) { best.w = v.w; iw = c; }
    }
    const int mx = smap[ix], my = smap[iy], mz = smap[iz], mw = smap[iw];

    float4* op = (float4*)out;
    #pragma unroll
    for (int c = 0; c < N_CH; ++c) {
        float4 o;
        o.x = (mx == c) ? 1.0f : 0.0f;
        o.y = (my == c) ? 1.0f : 0.0f;
        o.z = (mz == c) ? 1.0f : 0.0f;
        o.w = (mw == c) ? 1.0f : 0.0f;
        op[c * PLANE4 + g] = o;
    }
}

// ---------------------------------------------------------------------------
// Host launcher
// ---------------------------------------------------------------------------
extern "C" void kernel_launch(void* const* d_in, const int* in_sizes, int n_in,
                              void* d_out, int out_size, void* d_ws, size_t ws_size,
                              hipStream_t stream) {
    const float* masks    = (const float*)d_in[0];   // (1,32,1024,1024) f32
    const float* sf       = (const float*)d_in[1];   // (4,7,64) f32
    const int*   pad_left = (const int*)d_in[2];     // (4,) i32
    const int*   pad_top  = (const int*)d_in[3];     // (4,) i32
    float* out = (float*)d_out;
    int*   ws  = (int*)d_ws;                         // needs 160 ints (640 B)

    mw_zero_kernel<<<1, 128, 0, stream>>>(ws);
    mw_edges_kernel<<<N_WIN, WIN_H, 0, stream>>>(masks, pad_left, pad_top, ws);
    mw_gram_merge_kernel<<<1, 32, 0, stream>>>(sf, pad_left, pad_top, ws, ws + WS_MAP);
    mw_onehot_kernel<<<PLANE4 / 256, 256, 0, stream>>>(masks, ws + WS_MAP, out);
}